// RGBModel_35218731828067
// MI455X (gfx1250) — compile-verified
//
#include <hip/hip_runtime.h>

// ---------------------------------------------------------------------------
// NeRF-style back-to-front alpha compositing, MI455X (gfx1250).
// Memory-bound: 512 MB in / 3 MB out @ 23.3 TB/s  => ~22 us floor.
// Strategy: stream input through LDS with CDNA5 async global->LDS copies
// (ASYNCcnt path), double-buffered 64KB tiles, conflict-free padded LDS rows,
// one ray per thread for the serial scan.
// ---------------------------------------------------------------------------

#define N_RAYS        262144
#define N_SAMPLES     128
#define RAYS_PER_BLK  256
#define CHUNK         16                       // samples per LDS tile
#define NCHUNKS       (N_SAMPLES / CHUNK)      // 8
#define RAY_BYTES     (N_SAMPLES * 16)         // 2048 B per ray in global
#define ROW_BYTES     (CHUNK * 16 + 16)        // 272 B: pad => conflict-free LDS
#define TILE_BYTES    (RAYS_PER_BLK * ROW_BYTES) // 69632 B
#define LDS_BYTES     (2 * TILE_BYTES)         // 139264 B (2 blocks / 320KB WGP)
#define LOADS_PER_CHUNK ((RAYS_PER_BLK * CHUNK) / RAYS_PER_BLK) // 16 per thread/wave

// Builtin signature (from hipcc diagnostic): first param is
// 'int __vector(4) __device__ *'  => v4i in AS1 (global); LDS side is AS3.
typedef int v4i __attribute__((vector_size(4 * sizeof(int))));
typedef __attribute__((address_space(1))) v4i* as1_v4i;
typedef __attribute__((address_space(3))) v4i* as3_v4i;
typedef __attribute__((address_space(3))) void* as3_ptr;

__device__ __forceinline__ void async_load_b128(const void* g, void* l) {
#if __has_builtin(__builtin_amdgcn_global_load_async_to_lds_b128)
    __builtin_amdgcn_global_load_async_to_lds_b128(
        (as1_v4i)(v4i*)(void*)g, (as3_v4i)(v4i*)l, /*offset=*/0, /*cpol=*/0);
#else
    unsigned loff = (unsigned)(unsigned long long)(as3_ptr)l;
    asm volatile("global_load_async_to_lds_b128 %0, %1, off"
                 :: "v"(loff), "v"((unsigned long long)g)
                 : "memory");
#endif
}

template <int N>
__device__ __forceinline__ void wait_asynccnt() {
#if __has_builtin(__builtin_amdgcn_s_wait_asynccnt)
    __builtin_amdgcn_s_wait_asynccnt(N);
#else
    asm volatile("s_wait_asynccnt %0" :: "n"(N) : "memory");
#endif
}

__global__ void __launch_bounds__(RAYS_PER_BLK)
composite_kernel(const float* __restrict__ src, float* __restrict__ out) {
    extern __shared__ char smem[];

    const int  tid        = threadIdx.x;                 // one ray per thread
    const long long blk_ray0 = (long long)blockIdx.x * RAYS_PER_BLK;
    const char* gbase = (const char*)src + blk_ray0 * RAY_BYTES;

    // Issue async copy of one 16-sample chunk for all 256 rays of this block.
    // Lane-consecutive e => 256B-contiguous global runs (coalesced).
    auto issue_chunk = [&](int c, int buf) {
        char* lbase = smem + buf * TILE_BYTES;
        const int s0 = c * CHUNK;
#pragma unroll
        for (int k = 0; k < LOADS_PER_CHUNK; ++k) {
            int e = tid + k * RAYS_PER_BLK;      // 0..4095
            int r = e >> 4;                      // ray in block
            int s = e & (CHUNK - 1);             // sample in chunk
            const char* g = gbase + (long long)r * RAY_BYTES
                                  + (long long)(s0 + s) * 16;
            char* l = lbase + r * ROW_BYTES + s * 16;
            async_load_b128(g, l);
        }
    };

    int p = 0;
    issue_chunk(NCHUNKS - 1, 0);                 // prologue: last chunk first

    float tr = 0.0f, tg = 0.0f, tb = 0.0f;       // running composite (r,g,b)

    for (int c = NCHUNKS - 1; c >= 0; --c) {
        if (c > 0) {
            issue_chunk(c - 1, p ^ 1);           // prefetch next chunk (16 ops)
            wait_asynccnt<LOADS_PER_CHUNK>();    // in-order => current chunk done
        } else {
            wait_asynccnt<0>();
        }
        __syncthreads();                         // all waves' copies visible

        const float4* row =
            (const float4*)(smem + p * TILE_BYTES + tid * ROW_BYTES);
#pragma unroll
        for (int s = CHUNK - 1; s >= 0; --s) {   // reverse scan within chunk
            float4 v   = row[s];                 // {alpha, r, g, b}
            float  oma = 1.0f - v.x;
            tr = fmaf(oma, tr, v.x * v.y);
            tg = fmaf(oma, tg, v.x * v.z);
            tb = fmaf(oma, tb, v.x * v.w);
        }
        __syncthreads();                         // done reading buf p before reuse
        p ^= 1;
    }

    float* o = out + (blk_ray0 + tid) * 3;
    o[0] = tr; o[1] = tg; o[2] = tb;
}

extern "C" void kernel_launch(void* const* d_in, const int* in_sizes, int n_in,
                              void* d_out, int out_size, void* d_ws, size_t ws_size,
                              hipStream_t stream) {
    (void)in_sizes; (void)n_in; (void)d_ws; (void)ws_size; (void)out_size;
    const float* src = (const float*)d_in[0];
    float*       out = (float*)d_out;

    dim3 grid(N_RAYS / RAYS_PER_BLK);            // 1024 blocks
    dim3 block(RAYS_PER_BLK);                    // 256 threads = 8 waves
    hipLaunchKernelGGL(composite_kernel, grid, block, LDS_BYTES, stream, src, out);
}